// GNNEdgeClassifier_36447092474276
// MI455X (gfx1250) — compile-verified
//
#include <hip/hip_runtime.h>
#include <hip/hip_bf16.h>

#define N_NODES 100000
#define N_EDGES 500000
#define NODE_IN 16
#define HID 128

typedef __attribute__((ext_vector_type(2))) float v2f;
typedef __attribute__((ext_vector_type(8))) float v8f;

__device__ __forceinline__ float elu_f(float v) {
    return v > 0.f ? v : (__expf(v) - 1.f);
}

// D = A(16x4 f32) * B(4x16 f32) + C(16x16 f32), exact fp32 WMMA on CDNA5.
__device__ __forceinline__ v8f wmma4(v2f a, v2f b, v8f c) {
    return __builtin_amdgcn_wmma_f32_16x16x4_f32(
        /*neg_a=*/false, a, /*neg_b=*/false, b,
        /*c_mod=*/(short)0, c, /*reuse_a=*/false, /*reuse_b=*/false);
}

__device__ __forceinline__ void atomAddF(float* p, float v) {
    unsafeAtomicAdd(p, v);   // global_atomic_add_f32
}

// ---------------------------------------------------------------------------
// Simple float4 copy: aggr <- h  (implements the "(1+eps)*x" term with eps=0)
// ---------------------------------------------------------------------------
__global__ void copy_f4(const float4* __restrict__ src, float4* __restrict__ dst, int n4) {
    int i = blockIdx.x * blockDim.x + threadIdx.x;
    if (i < n4) dst[i] = src[i];
}

// ---------------------------------------------------------------------------
// Layer-0 message scatter: 16 features/edge, 16 threads per edge.
// m = relu(x[src] + eattr*We + be); aggr[dst] += m
// ---------------------------------------------------------------------------
__global__ void scatter16(const float* __restrict__ x,
                          const int* __restrict__ srcI, const int* __restrict__ dstI,
                          const float* __restrict__ eattr,
                          const float* __restrict__ We, const float* __restrict__ be,
                          float* __restrict__ aggr) {
    int idx = blockIdx.x * blockDim.x + threadIdx.x;
    int e = idx >> 4;
    if (e >= N_EDGES) return;
    int f = idx & 15;
    int s = srcI[e], d = dstI[e];
    float ea = eattr[e];
    float m = x[(size_t)s * NODE_IN + f] + ea * We[f] + be[f];
    m = fmaxf(m, 0.f);
    atomAddF(&aggr[(size_t)d * NODE_IN + f], m);
}

// ---------------------------------------------------------------------------
// HID-width message scatter: 128 features/edge, 32 threads * float4 per edge.
// ---------------------------------------------------------------------------
__global__ void scatter128(const float* __restrict__ h,
                           const int* __restrict__ srcI, const int* __restrict__ dstI,
                           const float* __restrict__ eattr,
                           const float* __restrict__ We, const float* __restrict__ be,
                           float* __restrict__ aggr) {
    int idx = blockIdx.x * blockDim.x + threadIdx.x;
    int e = idx >> 5;
    if (e >= N_EDGES) return;
    int f = (idx & 31) * 4;
    int s = srcI[e], d = dstI[e];
    float ea = eattr[e];
    float4 hv = *(const float4*)(h + (size_t)s * HID + f);
    float4 wv = *(const float4*)(We + f);
    float4 bv = *(const float4*)(be + f);
    float m0 = fmaxf(hv.x + ea * wv.x + bv.x, 0.f);
    float m1 = fmaxf(hv.y + ea * wv.y + bv.y, 0.f);
    float m2 = fmaxf(hv.z + ea * wv.z + bv.z, 0.f);
    float m3 = fmaxf(hv.w + ea * wv.w + bv.w, 0.f);
    float* ap = aggr + (size_t)d * HID + f;
    atomAddF(ap + 0, m0);
    atomAddF(ap + 1, m1);
    atomAddF(ap + 2, m2);
    atomAddF(ap + 3, m3);
}

// ---------------------------------------------------------------------------
// 16-row-tile GEMM inner loop against W[K x HID], accumulating 2 column tiles
// per wave. A fragments come from LDS (16 x STRIDE row-major), B streamed from
// global (L2-resident weights). f32 WMMA layout: lane<16 -> K,K+1 ; lane>=16 ->
// K+2,K+3 of the same 4-wide K block.
// ---------------------------------------------------------------------------
template<int STRIDE>
__device__ __forceinline__ void tile_gemm(const float* lin, const float* __restrict__ W,
                                          v8f& acc0, v8f& acc1,
                                          int lrow, int kOff, int n0) {
    #pragma unroll 4
    for (int kb = 0; kb < STRIDE / 4; ++kb) {
        int k = kb * 4 + kOff;
        v2f a;
        a.x = lin[lrow * STRIDE + k];
        a.y = lin[lrow * STRIDE + k + 1];
        v2f bf0, bf1;
        bf0.x = W[(size_t)k * HID + n0 + lrow];
        bf0.y = W[(size_t)(k + 1) * HID + n0 + lrow];
        bf1.x = W[(size_t)k * HID + n0 + 16 + lrow];
        bf1.y = W[(size_t)(k + 1) * HID + n0 + 16 + lrow];
        acc0 = wmma4(a, bf0, acc0);
        acc1 = wmma4(a, bf1, acc1);
    }
}

// ---------------------------------------------------------------------------
// Per-node MLP: Hout = ELU( ELU(Hin @ W1 + b1) @ W2 + b2 )
// One block (4 wave32) per 16-node tile; each wave owns 32 output columns.
// ---------------------------------------------------------------------------
template<int K1>
__global__ void node_mlp(const float* __restrict__ Hin,    // [N, K1]
                         const float* __restrict__ W1,     // [K1, HID]
                         const float* __restrict__ bias1,  // [HID]
                         const float* __restrict__ W2,     // [HID, HID]
                         const float* __restrict__ bias2,  // [HID]
                         float* __restrict__ Hout) {       // [N, HID]
    __shared__ float lds_in[16 * K1];
    __shared__ float lds_t[16 * HID];
    const int tid = threadIdx.x;
    const int tile = blockIdx.x;

    {   // stage contiguous 16*K1 input floats
        const float4* s4 = (const float4*)(Hin + (size_t)tile * 16 * K1);
        float4* d4 = (float4*)lds_in;
        for (int i = tid; i < (16 * K1) / 4; i += 128) d4[i] = s4[i];
    }
    __syncthreads();

    const int lane = tid & 31;
    const int wave = tid >> 5;
    const int half = lane >> 4;
    const int lrow = lane & 15;
    const int kOff = half * 2;
    const int n0 = wave * 32;

    // GEMM1 + bias + ELU -> lds_t
    v8f a0 = {}, a1 = {};
    tile_gemm<K1>(lds_in, W1, a0, a1, lrow, kOff, n0);
    #pragma unroll
    for (int v = 0; v < 8; ++v) {
        int row = half ? (v + 8) : v;
        int c0 = n0 + lrow, c1 = n0 + 16 + lrow;
        lds_t[row * HID + c0] = elu_f(a0[v] + bias1[c0]);
        lds_t[row * HID + c1] = elu_f(a1[v] + bias1[c1]);
    }
    __syncthreads();

    // GEMM2 + bias + outer ELU -> global
    v8f c0v = {}, c1v = {};
    tile_gemm<HID>(lds_t, W2, c0v, c1v, lrow, kOff, n0);
    #pragma unroll
    for (int v = 0; v < 8; ++v) {
        int row = half ? (v + 8) : v;
        int c0 = n0 + lrow, c1 = n0 + 16 + lrow;
        size_t base = ((size_t)tile * 16 + row) * HID;
        Hout[base + c0] = elu_f(c0v[v] + bias2[c0]);
        Hout[base + c1] = elu_f(c1v[v] + bias2[c1]);
    }
}

// ---------------------------------------------------------------------------
// Edge predictor: out[e] = ELU([h_src|h_dst] @ Wp1 + bp1) @ Wp2 + bp2
// One block per 16-edge tile; ef tile is 16x256 in LDS (K=256 WMMA GEMM).
// ---------------------------------------------------------------------------
__global__ void edge_pred(const float* __restrict__ h,
                          const int* __restrict__ srcI, const int* __restrict__ dstI,
                          const float* __restrict__ Wp1,   // [256, HID]
                          const float* __restrict__ bp1,   // [HID]
                          const float* __restrict__ Wp2,   // [HID]
                          const float* __restrict__ bp2,   // [1]
                          float* __restrict__ out) {       // [E]
    __shared__ float ef[16 * 256];
    __shared__ float tt[16 * HID];
    const int tid = threadIdx.x;
    const int tile = blockIdx.x;

    {   // gather [h_src | h_dst]; 8 threads per edge row
        int r = tid >> 3, part = tid & 7;
        int e = tile * 16 + r;
        int s = srcI[e], d = dstI[e];
        const float4* hs = (const float4*)(h + (size_t)s * HID);
        const float4* hd = (const float4*)(h + (size_t)d * HID);
        float4* efr = (float4*)(ef + r * 256);
        #pragma unroll
        for (int j = 0; j < 4; ++j) {
            int c4 = part * 4 + j;      // 0..31 float4 chunks (128 floats)
            efr[c4] = hs[c4];
            efr[32 + c4] = hd[c4];
        }
    }
    __syncthreads();

    const int lane = tid & 31;
    const int wave = tid >> 5;
    const int half = lane >> 4;
    const int lrow = lane & 15;
    const int kOff = half * 2;
    const int n0 = wave * 32;

    v8f a0 = {}, a1 = {};
    tile_gemm<256>(ef, Wp1, a0, a1, lrow, kOff, n0);
    #pragma unroll
    for (int v = 0; v < 8; ++v) {
        int row = half ? (v + 8) : v;
        int c0 = n0 + lrow, c1 = n0 + 16 + lrow;
        tt[row * HID + c0] = elu_f(a0[v] + bp1[c0]);
        tt[row * HID + c1] = elu_f(a1[v] + bp1[c1]);
    }
    __syncthreads();

    if (tid < 16) {   // final 128-dot against Wp2
        float sum = bp2[0];
        #pragma unroll 8
        for (int k = 0; k < HID; ++k) sum += tt[tid * HID + k] * Wp2[k];
        out[tile * 16 + tid] = sum;
    }
}

// ---------------------------------------------------------------------------
extern "C" void kernel_launch(void* const* d_in, const int* in_sizes, int n_in,
                              void* d_out, int out_size, void* d_ws, size_t ws_size,
                              hipStream_t stream) {
    const float* x     = (const float*)d_in[0];
    const int*   ei    = (const int*)d_in[1];
    const float* eattr = (const float*)d_in[2];
    const float* We0   = (const float*)d_in[3];
    const float* be0   = (const float*)d_in[4];
    const float* W10   = (const float*)d_in[5];
    const float* b10   = (const float*)d_in[6];
    const float* W20   = (const float*)d_in[7];
    const float* b20   = (const float*)d_in[8];
    const float* We_s  = (const float*)d_in[9];
    const float* be_s  = (const float*)d_in[10];
    const float* W1_s  = (const float*)d_in[11];
    const float* b1_s  = (const float*)d_in[12];
    const float* W2_s  = (const float*)d_in[13];
    const float* b2_s  = (const float*)d_in[14];
    const float* Wp1   = (const float*)d_in[15];
    const float* bp1   = (const float*)d_in[16];
    const float* Wp2   = (const float*)d_in[17];
    const float* bp2   = (const float*)d_in[18];

    const int* src = ei;
    const int* dst = ei + N_EDGES;

    float* h_cur = (float*)d_ws;                       // N*HID
    float* aggr  = h_cur + (size_t)N_NODES * HID;      // N*HID (first N*16 used by layer 0 as [N,16])
    float* out   = (float*)d_out;

    // ---- Layer 0 (NODE_IN=16 -> HID) ----
    {
        int n4 = N_NODES * NODE_IN / 4;
        copy_f4<<<(n4 + 255) / 256, 256, 0, stream>>>((const float4*)x, (float4*)aggr, n4);
        int total = N_EDGES * 16;
        scatter16<<<(total + 255) / 256, 256, 0, stream>>>(x, src, dst, eattr, We0, be0, aggr);
        node_mlp<NODE_IN><<<N_NODES / 16, 128, 0, stream>>>(aggr, W10, b10, W20, b20, h_cur);
    }

    // ---- Layers 1..4 (HID -> HID) ----
    for (int l = 0; l < 4; ++l) {
        int n4 = N_NODES * HID / 4;
        copy_f4<<<(n4 + 255) / 256, 256, 0, stream>>>((const float4*)h_cur, (float4*)aggr, n4);
        int total = N_EDGES * 32;
        scatter128<<<(total + 255) / 256, 256, 0, stream>>>(
            h_cur, src, dst, eattr,
            We_s + (size_t)l * HID, be_s + (size_t)l * HID, aggr);
        node_mlp<HID><<<N_NODES / 16, 128, 0, stream>>>(
            aggr,
            W1_s + (size_t)l * HID * HID, b1_s + (size_t)l * HID,
            W2_s + (size_t)l * HID * HID, b2_s + (size_t)l * HID, h_cur);
    }

    // ---- Edge predictor ----
    edge_pred<<<N_EDGES / 16, 128, 0, stream>>>(h_cur, src, dst, Wp1, bp1, Wp2, bp2, out);
}